// MIABlock_24799141167776
// MI455X (gfx1250) — compile-verified
//
#include <hip/hip_runtime.h>
#include <hip/hip_bf16.h>
#include <math.h>

// MIA block for MI455X (gfx1250, wave32).
//  - fp32 WMMA (v_wmma_f32_16x16x4_f32) for all GEMMs (exact fp32 path)
//  - single online-softmax pass over the 256MB U1 tensor with NT loads
//    (HBM-bound, ~11us floor at 23.3 TB/s; NT keeps L2 clean for weights)

typedef float v2f __attribute__((ext_vector_type(2)));
typedef float v4f __attribute__((ext_vector_type(4)));
typedef float v8f __attribute__((ext_vector_type(8)));

#define NROWS 4096   // B*T
#define DIM   256    // D
#define DIM1  64     // D1
#define FDIM  1024   // F
#define TT    256    // T

// ---------------------------------------------------------------- LayerNorm
// One wave per row of 256 floats (8 per lane), shuffle reductions (wave32).
__global__ __launch_bounds__(256)
void ln_kernel(const float* __restrict__ X, const float* __restrict__ gamma,
               const float* __restrict__ beta, float* __restrict__ Y)
{
    const int row  = blockIdx.x * 8 + (threadIdx.x >> 5);
    const int lane = threadIdx.x & 31;
    const float* xr = X + (size_t)row * DIM + lane * 8;
    v4f v0 = *(const v4f*)(xr);
    v4f v1 = *(const v4f*)(xr + 4);

    float s = v0.x + v0.y + v0.z + v0.w + v1.x + v1.y + v1.z + v1.w;
    #pragma unroll
    for (int off = 16; off > 0; off >>= 1) s += __shfl_xor(s, off, 32);
    const float mu = s * (1.0f / DIM);

    float q = 0.0f;
    q += (v0.x - mu) * (v0.x - mu); q += (v0.y - mu) * (v0.y - mu);
    q += (v0.z - mu) * (v0.z - mu); q += (v0.w - mu) * (v0.w - mu);
    q += (v1.x - mu) * (v1.x - mu); q += (v1.y - mu) * (v1.y - mu);
    q += (v1.z - mu) * (v1.z - mu); q += (v1.w - mu) * (v1.w - mu);
    #pragma unroll
    for (int off = 16; off > 0; off >>= 1) q += __shfl_xor(q, off, 32);
    const float inv = rsqrtf(q * (1.0f / DIM) + 1e-5f);

    const float* g = gamma + lane * 8;
    const float* b = beta  + lane * 8;
    v4f o0, o1;
    o0.x = (v0.x - mu) * inv * g[0] + b[0];
    o0.y = (v0.y - mu) * inv * g[1] + b[1];
    o0.z = (v0.z - mu) * inv * g[2] + b[2];
    o0.w = (v0.w - mu) * inv * g[3] + b[3];
    o1.x = (v1.x - mu) * inv * g[4] + b[4];
    o1.y = (v1.y - mu) * inv * g[5] + b[5];
    o1.z = (v1.z - mu) * inv * g[6] + b[6];
    o1.w = (v1.w - mu) * inv * g[7] + b[7];
    float* yr = Y + (size_t)row * DIM + lane * 8;
    *(v4f*)(yr)     = o0;
    *(v4f*)(yr + 4) = o1;
}

// ---------------------------------------------------------------- WMMA GEMM
// C[.,N] = A[.,K] @ B[K,N] (+ epilogue), N and K compile-time so B-load
// strides become instruction-immediate offsets.
// Block tile 128(M) x 64(N), 8 waves in 4x2; wave tile 32x32 = 4 fp32 WMMA
// accumulators (4 independent XDL chains per k-step).
// A tile staged in LDS with pad-17 row stride (conflict-free for both lane
// halves and both M-fragments); B fragments read coalesced from global/L2.
#define GEMM_BK   16
#define GEMM_LDA  17

__device__ __forceinline__ float gelu_exact(float x) {
    return 0.5f * x * (1.0f + erff(x * 0.70710678118654752440f));
}

// EPI: 0=plain  1=+resid  2=+bias,gelu  3=+bias,+resid
template<int EPI, int N, int K>
__global__ __launch_bounds__(256)
void gemm_wmma_f32(const float* __restrict__ A, const float* __restrict__ B,
                   float* __restrict__ C, const float* __restrict__ resid,
                   const float* __restrict__ bias)
{
    __shared__ float As[128 * GEMM_LDA];   // 8704 B

    const int tid  = threadIdx.x;
    const int wave = tid >> 5;
    const int lane = tid & 31;
    const int lh   = lane >> 4;   // lane half (0/1)
    const int l15  = lane & 15;

    const int tileM = blockIdx.y * 128;
    const int tileN = blockIdx.x * 64;
    const int wm = (wave & 3) * 32;   // wave row offset in tile
    const int wn = (wave >> 2) * 32;  // wave col offset in tile

    v8f c00 = {}, c01 = {}, c10 = {}, c11 = {};   // [mfrag][nfrag]

    const int col0 = tileN + wn + l15;
    const int col1 = col0 + 16;

    // A staging: 128 rows x 16 cols, 8 floats per thread
    const int ldr = tid >> 1;
    const int ldc = (tid & 1) * 8;
    const float* aload = A + (size_t)(tileM + ldr) * K + ldc;

    // LDS fragment bases for this wave's two M-fragments
    const int arow0 = (wm + l15) * GEMM_LDA;
    const int arow1 = (wm + 16 + l15) * GEMM_LDA;

    for (int ks = 0; ks < K; ks += GEMM_BK) {
        __syncthreads();
        {
            const float* ap = aload + ks;
            v4f a0 = *(const v4f*)(ap);
            v4f a1 = *(const v4f*)(ap + 4);
            float* dst = &As[ldr * GEMM_LDA + ldc];
            dst[0] = a0.x; dst[1] = a0.y; dst[2] = a0.z; dst[3] = a0.w;
            dst[4] = a1.x; dst[5] = a1.y; dst[6] = a1.z; dst[7] = a1.w;
        }
        __syncthreads();

        #pragma unroll
        for (int kk = 0; kk < GEMM_BK; kk += 4) {
            const int ko = kk + 2 * lh;           // this lane-half's K pair
            v2f a0, a1;
            a0.x = As[arow0 + ko]; a0.y = As[arow0 + ko + 1];
            a1.x = As[arow1 + ko]; a1.y = As[arow1 + ko + 1];
            // B fragment rows ks+ko, ks+ko+1; N is compile-time -> immediates
            const float* bp = B + (size_t)(ks + ko) * N;
            v2f b0; b0.x = bp[col0]; b0.y = bp[N + col0];
            v2f b1; b1.x = bp[col1]; b1.y = bp[N + col1];
            c00 = __builtin_amdgcn_wmma_f32_16x16x4_f32(false, a0, false, b0,
                                                        (short)0, c00, false, false);
            c01 = __builtin_amdgcn_wmma_f32_16x16x4_f32(false, a0, false, b1,
                                                        (short)0, c01, false, false);
            c10 = __builtin_amdgcn_wmma_f32_16x16x4_f32(false, a1, false, b0,
                                                        (short)0, c10, false, false);
            c11 = __builtin_amdgcn_wmma_f32_16x16x4_f32(false, a1, false, b1,
                                                        (short)0, c11, false, false);
        }
    }

    #pragma unroll
    for (int i = 0; i < 8; ++i) {
        #pragma unroll
        for (int mf = 0; mf < 2; ++mf) {
            const int row = tileM + wm + mf * 16 + i + 8 * lh;
            const size_t base = (size_t)row * N;
            float v0 = (mf == 0) ? c00[i] : c10[i];
            float v1 = (mf == 0) ? c01[i] : c11[i];
            if constexpr (EPI == 2 || EPI == 3) { v0 += bias[col0]; v1 += bias[col1]; }
            if constexpr (EPI == 2)             { v0 = gelu_exact(v0); v1 = gelu_exact(v1); }
            if constexpr (EPI == 1 || EPI == 3) { v0 += resid[base + col0]; v1 += resid[base + col1]; }
            C[base + col0] = v0;
            C[base + col1] = v1;
        }
    }
}

// ------------------------------------------------- online softmax attention
// mia[b,i,c] = sum_j softmax_j(U1[b,i,j,c] | mask[b,j]) * V[b,j,c]
// One block per (b,i). Thread t owns 4 channels (cq) x 16 source tokens (jg).
// Single pass over U1 (256 MB) with running (max, sum, weighted-acc); U1 read
// non-temporally so the one-shot stream doesn't wash the 192MB L2.
#define NEG_BIG (-3.0e38f)

__device__ __forceinline__ void online_upd(float u, float v,
                                           float& m, float& s, float& a) {
    if (u <= m) {
        float e = __expf(u - m);
        s += e;
        a += e * v;
    } else {
        float r = __expf(m - u);   // underflows to 0 when m == NEG_BIG
        s = s * r + 1.0f;
        a = a * r + v;
        m = u;
    }
}

__device__ __forceinline__ void merge_upd(float m2, float s2, float a2,
                                          float& m, float& s, float& a) {
    float mn = fmaxf(m, m2);
    float r1 = (m  == mn) ? 1.0f : __expf(m  - mn);
    float r2 = (m2 == mn) ? 1.0f : __expf(m2 - mn);
    s = s * r1 + s2 * r2;
    a = a * r1 + a2 * r2;
    m = mn;
}

__global__ __launch_bounds__(256)
void mia_attn_kernel(const float* __restrict__ U1, const float* __restrict__ V,
                     const unsigned char* __restrict__ mask,
                     float* __restrict__ mia)
{
    const int bi = blockIdx.x;          // b*T + i
    const int b  = bi >> 8;             // T = 256
    const int tid = threadIdx.x;
    const int cq = (tid & 15) * 4;      // channel quad base: 0..60
    const int jg = tid >> 4;            // j-group: 0..15 (16 tokens each)

    const float* u1r = U1 + (size_t)bi * (TT * DIM1);
    const float* vb  = V  + (size_t)b  * (TT * DIM1);

    // hoist this thread's 16 mask bytes into registers (4B aligned: j0 % 16 == 0)
    const int j0 = jg * 16;
    const unsigned int* mw = (const unsigned int*)(mask + b * TT + j0);
    unsigned int mk0 = mw[0], mk1 = mw[1], mk2 = mw[2], mk3 = mw[3];

    float4 m = { NEG_BIG, NEG_BIG, NEG_BIG, NEG_BIG };
    float4 s = { 0.f, 0.f, 0.f, 0.f };
    float4 a = { 0.f, 0.f, 0.f, 0.f };

    #pragma unroll 4
    for (int jj = 0; jj < 16; ++jj) {
        const unsigned int mword = (jj < 4) ? mk0 : (jj < 8) ? mk1 : (jj < 12) ? mk2 : mk3;
        if (!((mword >> ((jj & 3) * 8)) & 0xffu)) continue;
        const size_t off = (size_t)(j0 + jj) * DIM1 + cq;
        v4f u  = __builtin_nontemporal_load((const v4f*)(u1r + off));
        v4f vv = *(const v4f*)(vb + off);
        online_upd(u.x, vv.x, m.x, s.x, a.x);
        online_upd(u.y, vv.y, m.y, s.y, a.y);
        online_upd(u.z, vv.z, m.z, s.z, a.z);
        online_upd(u.w, vv.w, m.w, s.w, a.w);
    }

    __shared__ float4 sm[16][16], ss[16][16], sa[16][16];  // 12 KB
    sm[jg][tid & 15] = m;
    ss[jg][tid & 15] = s;
    sa[jg][tid & 15] = a;
    __syncthreads();

    if (tid < 16) {
        float4 M = sm[0][tid], S = ss[0][tid], A = sa[0][tid];
        #pragma unroll
        for (int g = 1; g < 16; ++g) {
            float4 m2 = sm[g][tid], s2 = ss[g][tid], a2 = sa[g][tid];
            merge_upd(m2.x, s2.x, a2.x, M.x, S.x, A.x);
            merge_upd(m2.y, s2.y, a2.y, M.y, S.y, A.y);
            merge_upd(m2.z, s2.z, a2.z, M.z, S.z, A.z);
            merge_upd(m2.w, s2.w, a2.w, M.w, S.w, A.w);
        }
        v4f o;
        o.x = A.x / S.x; o.y = A.y / S.y; o.z = A.z / S.z; o.w = A.w / S.w;
        *(v4f*)(mia + (size_t)bi * DIM1 + tid * 4) = o;
    }
}

// ------------------------------------------------------------------ launch
extern "C" void kernel_launch(void* const* d_in, const int* in_sizes, int n_in,
                              void* d_out, int out_size, void* d_ws, size_t ws_size,
                              hipStream_t stream)
{
    const float* x        = (const float*)d_in[0];
    const float* U1       = (const float*)d_in[1];
    const unsigned char* mask = (const unsigned char*)d_in[2];  // jnp bool -> 1 byte
    const float* ln_mia_g = (const float*)d_in[3];
    const float* ln_mia_b = (const float*)d_in[4];
    const float* Wv       = (const float*)d_in[5];
    const float* Wo       = (const float*)d_in[6];
    const float* ln_ffn_g = (const float*)d_in[7];
    const float* ln_ffn_b = (const float*)d_in[8];
    const float* W1       = (const float*)d_in[9];
    const float* b1       = (const float*)d_in[10];
    const float* W2       = (const float*)d_in[11];
    const float* b2       = (const float*)d_in[12];
    float* out = (float*)d_out;

    // workspace carve-up (floats): 7.5M floats = 30 MB
    float* xn  = (float*)d_ws;                 // [4096,256]
    float* Vw  = xn  + (size_t)NROWS * DIM;    // [4096,64]
    float* mia = Vw  + (size_t)NROWS * DIM1;   // [4096,64]
    float* x2  = mia + (size_t)NROWS * DIM1;   // [4096,256]
    float* hn  = x2  + (size_t)NROWS * DIM;    // [4096,256]
    float* h1  = hn  + (size_t)NROWS * DIM;    // [4096,1024]

    // 1) xn = LN(x)
    ln_kernel<<<NROWS / 8, 256, 0, stream>>>(x, ln_mia_g, ln_mia_b, xn);

    // 2) V = xn @ Wv          [4096,256]x[256,64]
    gemm_wmma_f32<0, DIM1, DIM><<<dim3(DIM1 / 64, NROWS / 128), 256, 0, stream>>>(
        xn, Wv, Vw, nullptr, nullptr);

    // 3) mia = softmax_j(U1) . V   (single NT pass over 256 MB U1)
    mia_attn_kernel<<<NROWS, 256, 0, stream>>>(U1, Vw, mask, mia);

    // 4) x2 = x + mia @ Wo    [4096,64]x[64,256]
    gemm_wmma_f32<1, DIM, DIM1><<<dim3(DIM / 64, NROWS / 128), 256, 0, stream>>>(
        mia, Wo, x2, x, nullptr);

    // 5) hn = LN(x2)
    ln_kernel<<<NROWS / 8, 256, 0, stream>>>(x2, ln_ffn_g, ln_ffn_b, hn);

    // 6) h1 = gelu(hn @ W1 + b1)   [4096,256]x[256,1024]
    gemm_wmma_f32<2, FDIM, DIM><<<dim3(FDIM / 64, NROWS / 128), 256, 0, stream>>>(
        hn, W1, h1, nullptr, b1);

    // 7) out = x2 + h1 @ W2 + b2   [4096,1024]x[1024,256]
    gemm_wmma_f32<3, DIM, FDIM><<<dim3(DIM / 64, NROWS / 128), 256, 0, stream>>>(
        h1, W2, out, x2, b2);
}